// adapter3_19610820674011
// MI455X (gfx1250) — compile-verified
//
#include <hip/hip_runtime.h>
#include <math.h>

// ---------------- problem constants (fixed shapes from the reference) ------
#define DM      768      // D_MODEL
#define RR      192      // R
#define DI      384      // D_INNER
#define DSTATE  16
#define DCONV   4
#define DTRANK  12
#define NBATCH  2
#define LSEQ    4096     // d*h*w = 16*16*16
#define TOK     8192     // NBATCH * LSEQ
#define NCHUNK  16
#define CT      256      // LSEQ / NCHUNK

typedef float v2f __attribute__((ext_vector_type(2)));
typedef float v8f __attribute__((ext_vector_type(8)));

// ---------------------------------------------------------------------------
// fp32 WMMA GEMM, register-blocked 2x2:
//   C[M,N] = act( A[M,K] @ Bw[N,K]^T + bias ) + addend
// One wave (32 threads) per 32x32 output block; K-loop in steps of 4 using
// V_WMMA_F32_16X16X4_F32.  Per K-step: 4 b64 loads -> 4 WMMAs (1 load/WMMA).
// A-frag layout: lane m = lane&15, k = k0 + v + 2*(lane>=16), so each lane's
// two K values are adjacent in memory => single b64 load per fragment.
// act: 0 = none, 1 = softplus.  Requires M%32==0, N%32==0, K%4==0.
// ---------------------------------------------------------------------------
__device__ __forceinline__ v8f wmma4(v2f a, v2f b, v8f c)
{
    return __builtin_amdgcn_wmma_f32_16x16x4_f32(
        /*neg_a=*/false, a, /*neg_b=*/false, b,
        /*c_mod=*/(short)0, c, /*reuse_a=*/false, /*reuse_b=*/false);
}

__device__ __forceinline__ void store_tile(float* __restrict__ C, v8f acc,
                                           int mrow0, int ncol, int N,
                                           const float* __restrict__ bias,
                                           const float* __restrict__ addend,
                                           int act)
{
    const float bia = bias ? bias[ncol] : 0.0f;
#pragma unroll
    for (int v = 0; v < 8; ++v) {
        const int mrow = mrow0 + v;
        float val = acc[v] + bia;
        if (act == 1) val = (val > 20.0f) ? val : log1pf(__expf(val));  // softplus
        if (addend) val += addend[(size_t)mrow * N + ncol];
        C[(size_t)mrow * N + ncol] = val;
    }
}

__global__ void gemm_wmma_f32(const float* __restrict__ A,
                              const float* __restrict__ Bw,
                              const float* __restrict__ bias,
                              const float* __restrict__ addend,
                              float* __restrict__ C,
                              int M, int N, int K, int act)
{
    const int tileN = blockIdx.x * 32;
    const int tileM = blockIdx.y * 32;
    const int lane  = threadIdx.x & 31;
    const int half  = lane >> 4;      // 0: lanes 0-15, 1: lanes 16-31
    const int l16   = lane & 15;

    const float* __restrict__ arow0 = A  + (size_t)(tileM + l16) * K + 2 * half;
    const float* __restrict__ arow1 = arow0 + (size_t)16 * K;
    const float* __restrict__ brow0 = Bw + (size_t)(tileN + l16) * K + 2 * half;
    const float* __restrict__ brow1 = brow0 + (size_t)16 * K;

    v8f acc00 = {}, acc01 = {}, acc10 = {}, acc11 = {};
    for (int k0 = 0; k0 < K; k0 += 4) {
        __builtin_prefetch(arow0 + k0 + 64, 0, 1);   // global_prefetch_b8
        __builtin_prefetch(brow0 + k0 + 64, 0, 1);
        const float2 a0v = *(const float2*)(arow0 + k0);
        const float2 a1v = *(const float2*)(arow1 + k0);
        const float2 b0v = *(const float2*)(brow0 + k0);
        const float2 b1v = *(const float2*)(brow1 + k0);
        const v2f a0 = {a0v.x, a0v.y};
        const v2f a1 = {a1v.x, a1v.y};
        const v2f b0 = {b0v.x, b0v.y};
        const v2f b1 = {b1v.x, b1v.y};
        acc00 = wmma4(a0, b0, acc00);
        acc01 = wmma4(a0, b1, acc01);
        acc10 = wmma4(a1, b0, acc10);
        acc11 = wmma4(a1, b1, acc11);
    }

    const int ncol0 = tileN + l16;
    const int mrowA = tileM + 8 * half;        // subtile M=0..15
    const int mrowB = tileM + 16 + 8 * half;   // subtile M=16..31
    store_tile(C, acc00, mrowA, ncol0,      N, bias, addend, act);
    store_tile(C, acc01, mrowA, ncol0 + 16, N, bias, addend, act);
    store_tile(C, acc10, mrowB, ncol0,      N, bias, addend, act);
    store_tile(C, acc11, mrowB, ncol0 + 16, N, bias, addend, act);
}

// ---------------------------------------------------------------------------
// W_delta[c][j] = sum_r dt_proj_w[c][r] * x_proj_w[r][j]   (384x384, r<12)
// ---------------------------------------------------------------------------
__global__ void wdelta_kernel(const float* __restrict__ dt_proj_w,
                              const float* __restrict__ x_proj_w,
                              float* __restrict__ Wd)
{
    int idx = blockIdx.x * blockDim.x + threadIdx.x;
    if (idx >= DI * DI) return;
    int c = idx / DI, j = idx % DI;
    float s = 0.0f;
#pragma unroll
    for (int r = 0; r < DTRANK; ++r)
        s += dt_proj_w[c * DTRANK + r] * x_proj_w[r * DI + j];
    Wd[idx] = s;
}

// ---------------------------------------------------------------------------
// Causal depthwise conv(4) + SiLU.  xi = first DI cols of xz (row stride 768).
// ---------------------------------------------------------------------------
__global__ void conv_silu_kernel(const float* __restrict__ xz,
                                 const float* __restrict__ conv_w,
                                 const float* __restrict__ conv_b,
                                 float* __restrict__ u)
{
    int idx = blockIdx.x * blockDim.x + threadIdx.x;
    if (idx >= TOK * DI) return;
    int c   = idx % DI;
    int tok = idx / DI;
    int t    = tok & (LSEQ - 1);
    int base = tok - t;
    float acc = conv_b[c];
#pragma unroll
    for (int k = 0; k < DCONV; ++k) {
        int tt = t - (DCONV - 1) + k;
        if (tt >= 0)
            acc += xz[(size_t)(base + tt) * (2 * DI) + c] * conv_w[c * DCONV + k];
    }
    u[idx] = acc / (1.0f + __expf(-acc));   // silu
}

// ---------------------------------------------------------------------------
// Scan pass A: per-chunk (P = prod a_t, S = local scan with h0=0).
// lane = (channel-sub, state): lanes 0-15 = states of channel c, 16-31 = c+1.
// block 256 threads -> 16 channels; grid = (DI/16, NCHUNK, NBATCH).
// ---------------------------------------------------------------------------
__global__ void scan_chunk_stats(const float* __restrict__ delta,
                                 const float* __restrict__ u,
                                 const float* __restrict__ bc,
                                 const float* __restrict__ A_log,
                                 float* __restrict__ Pout,
                                 float* __restrict__ Sout)
{
    const int tid   = threadIdx.x;
    const int n     = tid & 15;
    const int c     = blockIdx.x * 16 + (tid >> 4);
    const int chunk = blockIdx.y;
    const int bb    = blockIdx.z;

    const float Aneg = -__expf(A_log[c * DSTATE + n]);
    float P = 1.0f, S = 0.0f;
    const int t0 = chunk * CT;
    for (int t = t0; t < t0 + CT; ++t) {
        const size_t tok = (size_t)bb * LSEQ + t;
        const float dv = delta[tok * DI + c];
        const float uv = u[tok * DI + c];
        const float Bv = bc[tok * 32 + n];
        const float a  = __expf(dv * Aneg);
        P = P * a;
        S = a * S + dv * uv * Bv;
    }
    const size_t o = (((size_t)bb * NCHUNK + chunk) * DI + c) * DSTATE + n;
    Pout[o] = P;
    Sout[o] = S;
}

// ---------------------------------------------------------------------------
// Scan pass B: exclusive prefix across 16 chunks per (b,c,n).
// ---------------------------------------------------------------------------
__global__ void scan_chunk_prefix(const float* __restrict__ P,
                                  const float* __restrict__ S,
                                  float* __restrict__ H0)
{
    int idx = blockIdx.x * blockDim.x + threadIdx.x;   // NBATCH*DI*16 = 12288
    if (idx >= NBATCH * DI * DSTATE) return;
    int bb = idx / (DI * DSTATE);
    int cn = idx % (DI * DSTATE);
    float h = 0.0f;
#pragma unroll
    for (int j = 0; j < NCHUNK; ++j) {
        size_t o = ((size_t)bb * NCHUNK + j) * (DI * DSTATE) + cn;
        H0[o] = h;
        h = P[o] * h + S[o];
    }
}

// ---------------------------------------------------------------------------
// Scan pass C: replay chunk with correct h0, reduce 16 states (shfl_xor
// within half-wave), fuse + u*D and * silu(z); write y (TOK x DI).
// ---------------------------------------------------------------------------
__global__ void scan_final(const float* __restrict__ delta,
                           const float* __restrict__ u,
                           const float* __restrict__ bc,
                           const float* __restrict__ A_log,
                           const float* __restrict__ H0,
                           const float* __restrict__ xz,   // z = cols [DI,2*DI)
                           const float* __restrict__ D_ssm,
                           float* __restrict__ y)
{
    const int tid   = threadIdx.x;
    const int n     = tid & 15;
    const int c     = blockIdx.x * 16 + (tid >> 4);
    const int chunk = blockIdx.y;
    const int bb    = blockIdx.z;

    const float Aneg = -__expf(A_log[c * DSTATE + n]);
    const float Dc   = D_ssm[c];
    float h = H0[(((size_t)bb * NCHUNK + chunk) * DI + c) * DSTATE + n];

    const int t0 = chunk * CT;
    for (int t = t0; t < t0 + CT; ++t) {
        const size_t tok = (size_t)bb * LSEQ + t;
        const float dv = delta[tok * DI + c];
        const float uv = u[tok * DI + c];
        const float Bv = bc[tok * 32 + n];
        const float Cv = bc[tok * 32 + 16 + n];
        const float a  = __expf(dv * Aneg);
        h = a * h + dv * uv * Bv;

        float partial = h * Cv;                    // reduce over 16 states
        partial += __shfl_xor(partial, 1, 32);
        partial += __shfl_xor(partial, 2, 32);
        partial += __shfl_xor(partial, 4, 32);
        partial += __shfl_xor(partial, 8, 32);

        if (n == 0) {
            float yv = partial + uv * Dc;
            const float zv = xz[tok * (2 * DI) + DI + c];
            yv *= zv / (1.0f + __expf(-zv));       // * silu(z)
            y[tok * DI + c] = yv;
        }
    }
}

// ---------------------------------------------------------------------------
// LayerNorm(192) + exact GeLU, one wave per row (6 cols/lane), in place.
// ---------------------------------------------------------------------------
__global__ void ln_gelu_kernel(float* __restrict__ m,
                               const float* __restrict__ g,
                               const float* __restrict__ bta)
{
    const int row  = blockIdx.x;
    const int lane = threadIdx.x;     // 32
    const size_t base = (size_t)row * RR;

    float vals[6];
    float s = 0.0f, s2 = 0.0f;
#pragma unroll
    for (int i = 0; i < 6; ++i) {
        float v = m[base + lane + 32 * i];
        vals[i] = v; s += v; s2 += v * v;
    }
#pragma unroll
    for (int off = 16; off; off >>= 1) {
        s  += __shfl_xor(s,  off, 32);
        s2 += __shfl_xor(s2, off, 32);
    }
    const float mean = s * (1.0f / RR);
    const float var  = s2 * (1.0f / RR) - mean * mean;
    const float rstd = rsqrtf(var + 1e-5f);
#pragma unroll
    for (int i = 0; i < 6; ++i) {
        const int col = lane + 32 * i;
        float v = (vals[i] - mean) * rstd * g[col] + bta[col];
        v = 0.5f * v * (1.0f + erff(v * 0.70710678118654752f));  // exact gelu
        m[base + col] = v;
    }
}

// ---------------------------------------------------------------------------
static inline void launch_gemm(const float* A, const float* Bw, const float* bias,
                               const float* addend, float* C,
                               int M, int N, int K, int act, hipStream_t s)
{
    dim3 grid(N / 32, M / 32);
    gemm_wmma_f32<<<grid, 32, 0, s>>>(A, Bw, bias, addend, C, M, N, K, act);
}

extern "C" void kernel_launch(void* const* d_in, const int* in_sizes, int n_in,
                              void* d_out, int out_size, void* d_ws, size_t ws_size,
                              hipStream_t stream)
{
    const float* x         = (const float*)d_in[0];   // (8192, 768)
    const float* down_w    = (const float*)d_in[1];   // (192, 768)
    const float* up_w      = (const float*)d_in[2];   // (768, 192)
    const float* in_proj_w = (const float*)d_in[3];   // (768, 192)
    const float* conv_w    = (const float*)d_in[4];   // (384, 4)
    const float* conv_b    = (const float*)d_in[5];   // (384,)
    const float* x_proj_w  = (const float*)d_in[6];   // (44, 384)
    const float* dt_proj_w = (const float*)d_in[7];   // (384, 12)
    const float* dt_proj_b = (const float*)d_in[8];   // (384,)
    const float* A_log     = (const float*)d_in[9];   // (384, 16)
    const float* D_ssm     = (const float*)d_in[10];  // (384,)
    const float* out_projw = (const float*)d_in[11];  // (192, 384)
    const float* ln_g      = (const float*)d_in[12];  // (192,)
    const float* ln_b      = (const float*)d_in[13];  // (192,)
    float* out = (float*)d_out;                       // (8192, 768)

    // ---- workspace layout (floats) ----
    float* ws = (float*)d_ws;
    size_t o = 0;
    float* xd    = ws + o; o += (size_t)TOK * RR;            // 8192x192
    float* xz    = ws + o; o += (size_t)TOK * 2 * DI;        // 8192x768
    float* u     = ws + o; o += (size_t)TOK * DI;            // 8192x384
    float* bcbuf = ws + o; o += (size_t)TOK * 32;            // 8192x32  (B|C)
    float* delta = ws + o; o += (size_t)TOK * DI;            // 8192x384
    float* Wd    = ws + o; o += (size_t)DI * DI;             // 384x384
    float* Pbuf  = ws + o; o += (size_t)NBATCH * NCHUNK * DI * DSTATE;
    float* Sbuf  = ws + o; o += (size_t)NBATCH * NCHUNK * DI * DSTATE;
    float* H0    = ws + o; o += (size_t)NBATCH * NCHUNK * DI * DSTATE;
    float* y     = ws + o; o += (size_t)TOK * DI;            // 8192x384
    float* mbuf  = xd;      // xd is dead after the in_proj GEMM -> reuse

    // 0) composed dt matrix: Wd = dt_proj_w @ x_proj_w[:12]
    wdelta_kernel<<<(DI * DI + 255) / 256, 256, 0, stream>>>(dt_proj_w, x_proj_w, Wd);

    // 1) xd = x @ down_w^T           (8192x768x192)
    launch_gemm(x, down_w, nullptr, nullptr, xd, TOK, RR, DM, 0, stream);

    // 2) xz = xd @ in_proj_w^T       (8192x192x768)
    launch_gemm(xd, in_proj_w, nullptr, nullptr, xz, TOK, 2 * DI, RR, 0, stream);

    // 3) u = silu(causal_conv4(xi))
    conv_silu_kernel<<<(TOK * DI + 255) / 256, 256, 0, stream>>>(xz, conv_w, conv_b, u);

    // 4) bc = u @ x_proj_w[12:44]^T  (8192x384x32)  -> [Bm | Cm]
    launch_gemm(u, x_proj_w + (size_t)DTRANK * DI, nullptr, nullptr,
                bcbuf, TOK, 32, DI, 0, stream);

    // 5) delta = softplus(u @ Wd^T + dt_proj_b)  (8192x384x384)
    launch_gemm(u, Wd, dt_proj_b, nullptr, delta, TOK, DI, DI, 1, stream);

    // 6-8) chunked parallel scan
    dim3 sg(DI / 16, NCHUNK, NBATCH);
    scan_chunk_stats<<<sg, 256, 0, stream>>>(delta, u, bcbuf, A_log, Pbuf, Sbuf);
    scan_chunk_prefix<<<(NBATCH * DI * DSTATE + 255) / 256, 256, 0, stream>>>(Pbuf, Sbuf, H0);
    scan_final<<<sg, 256, 0, stream>>>(delta, u, bcbuf, A_log, H0, xz, D_ssm, y);

    // 9) m = y @ out_proj_w^T        (8192x384x192)
    launch_gemm(y, out_projw, nullptr, nullptr, mbuf, TOK, RR, DI, 0, stream);

    // 10) LayerNorm + GeLU in place
    ln_gelu_kernel<<<TOK, 32, 0, stream>>>(mbuf, ln_g, ln_b);

    // 11) out = x + m @ up_w^T       (8192x192x768, fused residual)
    launch_gemm(mbuf, up_w, nullptr, x, out, TOK, DM, RR, 0, stream);
}